// TRA_Attention_15281493639325
// MI455X (gfx1250) — compile-verified
//
#include <hip/hip_runtime.h>
#include <math.h>

#define BB    8
#define CC    64
#define HDIM  256
#define WDIM  256
#define HW    65536          // 2^16
#define NHEAD 8
#define CPH   8

typedef float v2f __attribute__((ext_vector_type(2)));
typedef float v8f __attribute__((ext_vector_type(8)));

// ---------------------------------------------------------------------------
// Kernel 1: fused gate+qkv 1x1 conv.  t1[b][o][n] = sum_c W[o][c] * x[b][c][n]
// o in [0,256): 0..63 = w_gate rows, 64..255 = w_qkv rows.
// WMMA f32 16x16x4: A = 16 out-ch x 4 in-ch, B = 4 in-ch x 16 positions.
// Wave tile: M=16, N=64 (4 sub-tiles reusing preloaded A). 8 waves/WG.
// ---------------------------------------------------------------------------
__global__ __launch_bounds__(256)
void k_conv1x1_gateqkv(const float* __restrict__ x,
                       const float* __restrict__ w_gate,
                       const float* __restrict__ w_qkv,
                       float* __restrict__ t1)
{
    const int lane  = threadIdx.x & 31;
    const int wave  = threadIdx.x >> 5;
    const int b     = blockIdx.z;
    const int mtile = blockIdx.y * 8 + wave;       // 0..15 (256 out ch)
    const int n0    = blockIdx.x * 64;

    const int mo   = mtile * 16;
    const int mrow = mo + (lane & 15);
    const int kh   = (lane >> 4) * 2;              // lanes 0-15: k{0,1}; 16-31: k{2,3}

    // Preload A (weights): 16 k-steps, contiguous pair per lane.
    const float* wrow = (mrow < CC) ? (w_gate + mrow * CC)
                                    : (w_qkv + (mrow - CC) * CC);
    v2f A[16];
#pragma unroll
    for (int ks = 0; ks < 16; ++ks)
        A[ks] = *(const v2f*)(wrow + ks * 4 + kh);

    const float* xb   = x + (size_t)b * CC * HW;
    const int    ncol = lane & 15;
    const int    rsel = 8 * (lane >> 4);

#pragma unroll
    for (int nt = 0; nt < 4; ++nt) {
        const int n = n0 + nt * 16 + ncol;
        v8f acc = {};
#pragma unroll
        for (int ks = 0; ks < 16; ++ks) {
            const int kb = ks * 4 + kh;
            v2f Bv;
            Bv.x = xb[(size_t)kb * HW + n];
            Bv.y = xb[(size_t)(kb + 1) * HW + n];
            acc = __builtin_amdgcn_wmma_f32_16x16x4_f32(
                      false, A[ks], false, Bv, (short)0, acc, false, false);
        }
        float* outp = t1 + ((size_t)b * 256 + mo + rsel) * HW + n;
#pragma unroll
        for (int v = 0; v < 8; ++v)
            outp[(size_t)v * HW] = acc[v];
    }
}

// ---------------------------------------------------------------------------
// Kernel 2: depthwise 3x3, pad 1, over 256 channels (gate + qkv).
// ---------------------------------------------------------------------------
__global__ __launch_bounds__(256)
void k_dwconv3x3(const float* __restrict__ t1,
                 const float* __restrict__ w_gate_dw,
                 const float* __restrict__ w_qkv_dw,
                 float* __restrict__ t2)
{
    const size_t idx = (size_t)blockIdx.x * 256 + threadIdx.x; // b*256*HW space
    const int    n   = (int)(idx & (HW - 1));
    const size_t bc  = idx >> 16;
    const int    ch  = (int)(bc & 255);
    const int    h   = n >> 8, w = n & 255;

    const float* wdw = (ch < CC) ? (w_gate_dw + ch * 9)
                                 : (w_qkv_dw + (ch - CC) * 9);
    const float* src = t1 + (bc << 16);

    float acc = 0.f;
#pragma unroll
    for (int dy = -1; dy <= 1; ++dy) {
        const int hh = h + dy;
        if (hh < 0 || hh > 255) continue;
#pragma unroll
        for (int dx = -1; dx <= 1; ++dx) {
            const int ww = w + dx;
            if (ww < 0 || ww > 255) continue;
            acc = fmaf(wdw[(dy + 1) * 3 + (dx + 1)], src[hh * 256 + ww], acc);
        }
    }
    t2[idx] = acc;
}

// ---------------------------------------------------------------------------
// Kernel 3: per-row sum-of-squares for q (t2 ch 64..127) and k (128..191).
// norms[r] = max(sqrt(sum), 1e-12), r = b*128 + cr  (cr<64: q, else k).
// ---------------------------------------------------------------------------
__global__ __launch_bounds__(256)
void k_rownorm(const float* __restrict__ t2, float* __restrict__ norms)
{
    __shared__ float red[256];
    const int r  = blockIdx.x;           // 0..1023
    const int b  = r >> 7;
    const int cr = r & 127;
    const float* row = t2 + ((size_t)b * 256 + 64 + cr) * HW;

    float s = 0.f;
    for (int i = threadIdx.x; i < HW; i += 256) {
        const float v = row[i];
        s = fmaf(v, v, s);
    }
    red[threadIdx.x] = s;
    __syncthreads();
    for (int off = 128; off > 0; off >>= 1) {
        if ((int)threadIdx.x < off) red[threadIdx.x] += red[threadIdx.x + off];
        __syncthreads();
    }
    if (threadIdx.x == 0) norms[r] = fmaxf(sqrtf(red[0]), 1e-12f);
}

// ---------------------------------------------------------------------------
// Zero the 16x16 attn accumulation tiles (global fp32 atomics target).
// ---------------------------------------------------------------------------
__global__ __launch_bounds__(256)
void k_zero(float* __restrict__ p, int n)
{
    const int i = blockIdx.x * 256 + threadIdx.x;
    if (i < n) p[i] = 0.f;
}

// ---------------------------------------------------------------------------
// Kernel 4: attn_raw[b,h,c,d] = sum_n q[c,n]*k[d,n]  via WMMA f32 16x16x4,
// 8x8 real data zero-padded to 16x16, K(=spatial) split 64 ways + atomics.
// A and B per-lane loads are contiguous v2f (K is the contiguous axis).
// ---------------------------------------------------------------------------
__global__ __launch_bounds__(32)
void k_attn_qk(const float* __restrict__ t2, float* __restrict__ attn_raw)
{
    const int lane  = threadIdx.x;
    const int kspl  = blockIdx.x;        // 0..63
    const int h     = blockIdx.y;
    const int b     = blockIdx.z;
    const int hi    = lane >> 4;
    const int m     = lane & 15;         // A row == B column
    const int koff  = hi * 2;

    const bool   ok   = (m < CPH);
    const float* qrow = t2 + ((size_t)b * 256 + 64  + h * CPH + (ok ? m : 0)) * HW;
    const float* krow = t2 + ((size_t)b * 256 + 128 + h * CPH + (ok ? m : 0)) * HW;
    const v2f    zz   = {0.f, 0.f};

    v8f acc = {};
    const int nb0 = kspl * (HW / 64);
    for (int nb = nb0; nb < nb0 + (HW / 64); nb += 4) {
        const v2f a  = ok ? *(const v2f*)(qrow + nb + koff) : zz;
        const v2f bv = ok ? *(const v2f*)(krow + nb + koff) : zz;
        acc = __builtin_amdgcn_wmma_f32_16x16x4_f32(
                  false, a, false, bv, (short)0, acc, false, false);
    }

    float* ap = attn_raw + (size_t)(b * NHEAD + h) * 256;
#pragma unroll
    for (int v = 0; v < 8; ++v) {
        const int Mr = v + 8 * hi;
        if (Mr < CPH && m < CPH)
            atomicAdd(ap + Mr * 16 + m, acc[v]);
    }
}

// ---------------------------------------------------------------------------
// Kernel 5: fold in 1/(||q||*||k||) and temperature, softmax over d (8 wide).
// attn_s compact layout [b*8+h][8][8].
// ---------------------------------------------------------------------------
__global__ __launch_bounds__(256)
void k_softmax(const float* __restrict__ attn_raw,
               const float* __restrict__ norms,
               const float* __restrict__ temperature,
               float* __restrict__ attn_s)
{
    const int t = blockIdx.x * 256 + threadIdx.x;   // 0..511
    if (t >= BB * NHEAD * CPH) return;
    const int c  = t & 7;
    const int bh = t >> 3;
    const int h  = bh & 7, b = bh >> 3;

    const float* raw  = attn_raw + (size_t)bh * 256 + c * 16;
    const float  qn   = norms[b * 128 + h * CPH + c];
    const float  temp = temperature[h];

    float vals[8];
    float mx = -3.4e38f;
#pragma unroll
    for (int d = 0; d < 8; ++d) {
        const float kn = norms[b * 128 + 64 + h * CPH + d];
        const float v  = raw[d] / (qn * kn) * temp;
        vals[d] = v;
        mx = fmaxf(mx, v);
    }
    float s = 0.f;
#pragma unroll
    for (int d = 0; d < 8; ++d) { vals[d] = expf(vals[d] - mx); s += vals[d]; }
    const float inv = 1.f / s;
    float* o = attn_s + (size_t)bh * 64 + c * 8;
#pragma unroll
    for (int d = 0; d < 8; ++d) o[d] = vals[d] * inv;
}

// ---------------------------------------------------------------------------
// Kernel 6 (fused 6+7): per workgroup, for a 128-position block of batch b:
//   Phase 1: outg[c][n] = gelu(gate[c][n]) * sum_d attn[b,h,c%8,d] * v[d][n]
//            staged in LDS (row stride 144 floats -> conflict-free B reads).
//   Phase 2: 8 waves run the w_proj GEMM (M=64, K=64) with WMMA f32 16x16x4,
//            B fragments read from LDS, D written straight to d_out.
// Eliminates the 128 MB outg global round-trip.
// ---------------------------------------------------------------------------
#define LSTR 144   // LDS row stride in floats (128 data + 16 pad)

__global__ __launch_bounds__(256)
void k_attnv_gelu_proj(const float* __restrict__ t2,
                       const float* __restrict__ attn_s,
                       const float* __restrict__ w_proj,
                       float* __restrict__ out)
{
    __shared__ float ldsX[CC * LSTR];   // 64 ch x 128 pos (+pad) = 36 KB

    const int b  = blockIdx.z;
    const int n0 = blockIdx.x * 128;

    // ---- Phase 1: attn @ v, gelu(gate) multiply -> LDS --------------------
    {
        const int nl   = threadIdx.x & 127;       // position within block
        const int hgrp = threadIdx.x >> 7;        // 0: heads 0-3, 1: heads 4-7
        const int n    = n0 + nl;

#pragma unroll
        for (int hh = 0; hh < 4; ++hh) {
            const int h = hgrp * 4 + hh;
            const float* vbase = t2 + ((size_t)b * 256 + 192 + h * CPH) * HW + n;
            const float* gbase = t2 + ((size_t)b * 256 + h * CPH) * HW + n;
            const float* A     = attn_s + (size_t)(b * NHEAD + h) * 64;

            float vv[8];
#pragma unroll
            for (int d = 0; d < 8; ++d) vv[d] = vbase[(size_t)d * HW];

#pragma unroll
            for (int c = 0; c < 8; ++c) {
                float s = 0.f;
#pragma unroll
                for (int d = 0; d < 8; ++d) s = fmaf(A[c * 8 + d], vv[d], s);
                const float g    = gbase[(size_t)c * HW];
                const float gelu = 0.5f * g * (1.f + erff(g * 0.70710678118654752f));
                ldsX[(h * CPH + c) * LSTR + nl] = gelu * s;
            }
        }
    }
    __syncthreads();

    // ---- Phase 2: proj GEMM from LDS (WMMA f32 16x16x4) -------------------
    {
        const int lane  = threadIdx.x & 31;
        const int wave  = threadIdx.x >> 5;
        const int mtile = wave & 3;               // 16 out-ch per wave
        const int nhalf = wave >> 2;              // 64 positions per wave

        const int mo   = mtile * 16;
        const int mrow = mo + (lane & 15);
        const int kh   = (lane >> 4) * 2;

        const float* wrow = w_proj + mrow * CC;
        v2f A[16];
#pragma unroll
        for (int ks = 0; ks < 16; ++ks)
            A[ks] = *(const v2f*)(wrow + ks * 4 + kh);

        const int ncol = lane & 15;
        const int rsel = 8 * (lane >> 4);

#pragma unroll
        for (int nt = 0; nt < 4; ++nt) {
            const int nl = nhalf * 64 + nt * 16 + ncol;
            v8f acc = {};
#pragma unroll
            for (int ks = 0; ks < 16; ++ks) {
                const int kb = ks * 4 + kh;
                v2f Bv;
                Bv.x = ldsX[kb * LSTR + nl];
                Bv.y = ldsX[(kb + 1) * LSTR + nl];
                acc = __builtin_amdgcn_wmma_f32_16x16x4_f32(
                          false, A[ks], false, Bv, (short)0, acc, false, false);
            }
            float* outp = out + ((size_t)b * CC + mo + rsel) * HW + n0 + nl;
#pragma unroll
            for (int v = 0; v < 8; ++v)
                outp[(size_t)v * HW] = acc[v];
        }
    }
}

// ---------------------------------------------------------------------------
// Host-side launch: pipeline on `stream`, scratch carved from d_ws.
//   [0,   512MB) : t1  pre-depthwise  [B][256][HW]
//   [512MB,1GB)  : t2  post-depthwise [B][256][HW]
//   then: norms (4KB), attn_raw (64KB), attn_s (16KB)
// ---------------------------------------------------------------------------
extern "C" void kernel_launch(void* const* d_in, const int* in_sizes, int n_in,
                              void* d_out, int out_size, void* d_ws, size_t ws_size,
                              hipStream_t stream)
{
    const float* x         = (const float*)d_in[0];
    const float* w_gate    = (const float*)d_in[1];
    const float* w_gate_dw = (const float*)d_in[2];
    const float* w_qkv     = (const float*)d_in[3];
    const float* w_qkv_dw  = (const float*)d_in[4];
    const float* w_proj    = (const float*)d_in[5];
    const float* temp      = (const float*)d_in[6];
    float*       out       = (float*)d_out;

    char* ws = (char*)d_ws;
    const size_t SZ_T = (size_t)BB * 256 * HW * sizeof(float); // 512 MB
    float* t1       = (float*)(ws);
    float* t2       = (float*)(ws + SZ_T);
    float* norms    = (float*)(ws + 2 * SZ_T);
    float* attn_raw = (float*)(ws + 2 * SZ_T + 4096);
    float* attn_s   = (float*)(ws + 2 * SZ_T + 4096 + 65536);

    // 1) fused gate+qkv 1x1 conv (WMMA GEMM, M=256)
    k_conv1x1_gateqkv<<<dim3(HW / 64, 2, BB), 256, 0, stream>>>(x, w_gate, w_qkv, t1);

    // 2) depthwise 3x3 over all 256 channels
    k_dwconv3x3<<<dim3((unsigned)((size_t)BB * 256 * HW / 256)), 256, 0, stream>>>(
        t1, w_gate_dw, w_qkv_dw, t2);

    // 3) q/k row norms
    k_rownorm<<<dim3(BB * 128), 256, 0, stream>>>(t2, norms);

    // 4) q.k^T via WMMA (K-split + atomics); zero the accumulator first
    k_zero<<<dim3(64), 256, 0, stream>>>(attn_raw, BB * NHEAD * 256);
    k_attn_qk<<<dim3(64, NHEAD, BB), 32, 0, stream>>>(t2, attn_raw);

    // 5) scale + softmax
    k_softmax<<<dim3(2), 256, 0, stream>>>(attn_raw, norms, temp, attn_s);

    // 6+7 fused) attn @ v  *  gelu(gate)  ->  LDS  ->  proj GEMM -> d_out
    k_attnv_gelu_proj<<<dim3(HW / 128, 1, BB), 256, 0, stream>>>(
        t2, attn_s, w_proj, out);

    (void)in_sizes; (void)n_in; (void)out_size; (void)ws_size;
}